// GAT_31069793419930
// MI455X (gfx1250) — compile-verified
//
#include <hip/hip_runtime.h>
#include <hip/hip_bf16.h>

typedef float v2f __attribute__((ext_vector_type(2)));
typedef float v8f __attribute__((ext_vector_type(8)));

#define NEG_SLOPE 0.2f
#define GAT_EPS 1e-16f

// ---------------------------------------------------------------------------
// GEMM: D[M,128] = A[M,128] @ B[128,128], fp32 WMMA 16x16x4.
// One wave per 16-row M tile; 8 N-tiles of 16 columns; 32 k-steps of K=4.
// B (64KB) is L2/WGP$-resident, loaded as fragments directly.
// ---------------------------------------------------------------------------
__global__ __launch_bounds__(256) void gat_gemm128_f32_wmma(
    const float* __restrict__ A, const float* __restrict__ B,
    float* __restrict__ D, int M) {
  const int wave  = threadIdx.x >> 5;                 // 0..7
  const int lane  = threadIdx.x & 31;
  const int m0    = (blockIdx.x * 8 + wave) * 16;
  if (m0 >= M) return;                                // wave-uniform exit
  const int lrow  = lane & 15;
  const int khalf = (lane >> 4) * 2;                  // 0 or 2 (K sub-offset)

  int arow = m0 + lrow;
  if (arow >= M) arow = M - 1;                        // clamp (results masked at store)
  const float* ap = A + (size_t)arow * 128 + khalf;

  v8f acc[8] = {};
  for (int k = 0; k < 128; k += 4) {
    v2f a = *(const v2f*)(ap + k);                    // A[m][k+khalf .. +1]
#pragma unroll
    for (int t = 0; t < 8; ++t) {
      const float* bp = B + (size_t)(k + khalf) * 128 + t * 16 + lrow;
      v2f b;
      b.x = bp[0];                                    // B[k+khalf][n]
      b.y = bp[128];                                  // B[k+khalf+1][n]
      acc[t] = __builtin_amdgcn_wmma_f32_16x16x4_f32(
          false, a, false, b, (short)0, acc[t], false, false);
    }
  }
  // C/D layout: lanes 0-15 -> N=lane, M=r; lanes 16-31 -> N=lane-16, M=8+r
  const int rbase = m0 + ((lane >> 4) ? 8 : 0);
#pragma unroll
  for (int t = 0; t < 8; ++t) {
    const int col = t * 16 + lrow;
#pragma unroll
    for (int r = 0; r < 8; ++r) {
      const int row = rbase + r;
      if (row < M) D[(size_t)row * 128 + col] = acc[t][r];
    }
  }
}

// ---------------------------------------------------------------------------
// Per (node, head): a_s = <h, att_src>, a_d = <h, att_dst>; init m=-inf, s=0.
// ---------------------------------------------------------------------------
__global__ __launch_bounds__(256) void gat_attn_prep(
    const float* __restrict__ h, const float* __restrict__ att_s,
    const float* __restrict__ att_d, float* __restrict__ a_s,
    float* __restrict__ a_d, float* __restrict__ m, float* __restrict__ s,
    int N) {
  int i = blockIdx.x * blockDim.x + threadIdx.x;      // [0, N*4)
  if (i >= N * 4) return;
  const int n = i >> 2, hd = i & 3;
  const float* hp = h + (size_t)n * 128 + hd * 32;
  const float* as = att_s + hd * 32;
  const float* ad = att_d + hd * 32;
  float ss = 0.f, dd = 0.f;
#pragma unroll
  for (int c = 0; c < 32; ++c) {
    const float v = hp[c];
    ss += v * as[c];
    dd += v * ad[c];
  }
  a_s[i] = ss;
  a_d[i] = dd;
  m[i]   = -__builtin_inff();
  s[i]   = 0.f;
}

// Branchless float atomic-max (valid with -inf init): int order for >=0,
// reversed uint order for <0.
__device__ __forceinline__ void gat_atomic_max_f32(float* addr, float val) {
  if (val >= 0.f) atomicMax((int*)addr, __float_as_int(val));
  else            atomicMin((unsigned int*)addr, __float_as_uint(val));
}

// ---------------------------------------------------------------------------
// Edge pass A: e = leaky_relu(a_s[src]+a_d[dst]); store; segment atomic max.
// Virtual edges [E, E+N) are self loops.
// ---------------------------------------------------------------------------
__global__ __launch_bounds__(256) void gat_edge_logit_max(
    const int* __restrict__ esrc, const int* __restrict__ edst,
    const float* __restrict__ a_s, const float* __restrict__ a_d,
    float* __restrict__ elog, float* __restrict__ m, int E, int NE) {
  int e = blockIdx.x * blockDim.x + threadIdx.x;
  if (e >= NE) return;
  const int s = (e < E) ? esrc[e] : (e - E);
  const int d = (e < E) ? edst[e] : (e - E);
  const float4 av = *(const float4*)(a_s + (size_t)s * 4);
  const float4 bv = *(const float4*)(a_d + (size_t)d * 4);
  float l[4] = {av.x + bv.x, av.y + bv.y, av.z + bv.z, av.w + bv.w};
  float* mp = m + (size_t)d * 4;
  float4 lo;
#pragma unroll
  for (int hd = 0; hd < 4; ++hd) {
    float v = l[hd] > 0.f ? l[hd] : NEG_SLOPE * l[hd];
    ((float*)&lo)[hd] = v;
    gat_atomic_max_f32(mp + hd, v);
  }
  *(float4*)(elog + (size_t)e * 4) = lo;
}

// ---------------------------------------------------------------------------
// Edge pass B: p = exp(e - m[dst]); store in place; segment atomic sum.
// ---------------------------------------------------------------------------
__global__ __launch_bounds__(256) void gat_edge_exp_sum(
    const int* __restrict__ edst, float* __restrict__ elog,
    const float* __restrict__ m, float* __restrict__ s, int E, int NE) {
  int e = blockIdx.x * blockDim.x + threadIdx.x;
  if (e >= NE) return;
  const int d = (e < E) ? edst[e] : (e - E);
  float4 l  = *(const float4*)(elog + (size_t)e * 4);
  const float4 mv = *(const float4*)(m + (size_t)d * 4);
  float* sp = s + (size_t)d * 4;
  l.x = __expf(l.x - mv.x);
  l.y = __expf(l.y - mv.y);
  l.z = __expf(l.z - mv.z);
  l.w = __expf(l.w - mv.w);
  atomicAdd(sp + 0, l.x);
  atomicAdd(sp + 1, l.y);
  atomicAdd(sp + 2, l.z);
  atomicAdd(sp + 3, l.w);
  *(float4*)(elog + (size_t)e * 4) = l;
}

// ---------------------------------------------------------------------------
// Edge pass C: alpha = p/(s[dst]+eps); out[dst] += h[src]*alpha.
// One wave32 per edge; lane covers 4 channels (head = lane/8). All traffic
// L2-resident (full working set < 192MB).
// ---------------------------------------------------------------------------
__global__ __launch_bounds__(256) void gat_edge_aggregate(
    const int* __restrict__ esrc, const int* __restrict__ edst,
    const float* __restrict__ p, const float* __restrict__ s,
    const float* __restrict__ h, float* __restrict__ out, int E, int NE) {
  const int gw   = (int)((blockIdx.x * 256u + threadIdx.x) >> 5);  // edge id
  const int lane = threadIdx.x & 31;
  if (gw >= NE) return;
  const int src  = (gw < E) ? esrc[gw] : (gw - E);
  const int dst  = (gw < E) ? edst[gw] : (gw - E);
  const int head = lane >> 3;                          // 8 lanes per head
  const float alpha = p[(size_t)gw * 4 + head] /
                      (s[(size_t)dst * 4 + head] + GAT_EPS);
  const float4 hv = *(const float4*)(h + (size_t)src * 128 + lane * 4);
  float* op = out + (size_t)dst * 128 + lane * 4;
  atomicAdd(op + 0, hv.x * alpha);
  atomicAdd(op + 1, hv.y * alpha);
  atomicAdd(op + 2, hv.z * alpha);
  atomicAdd(op + 3, hv.w * alpha);
}

// out = elu(out + b1)  (in place, [N,128])
__global__ __launch_bounds__(256) void gat_elu_bias(
    float* __restrict__ h, const float* __restrict__ b, int total) {
  int i = blockIdx.x * blockDim.x + threadIdx.x;
  if (i >= total) return;
  const float v = h[i] + b[i & 127];
  h[i] = v > 0.f ? v : (__expf(v) - 1.f);
}

// out[n,c] = mean over 4 heads of out2[n,h,c] + b2[c]
__global__ __launch_bounds__(256) void gat_final_mean(
    const float* __restrict__ o, const float* __restrict__ b2,
    float* __restrict__ out, int N) {
  int i = blockIdx.x * blockDim.x + threadIdx.x;      // [0, N*32)
  if (i >= N * 32) return;
  const int n = i >> 5, c = i & 31;
  const float* op = o + (size_t)n * 128 + c;
  out[i] = 0.25f * (op[0] + op[32] + op[64] + op[96]) + b2[c];
}

// ---------------------------------------------------------------------------
static void gat_run_layer(const float* xin, const float* W,
                          const float* att_s, const float* att_d,
                          float* h, float* outb,
                          float* as, float* ad, float* mm, float* sb,
                          float* elog,
                          const int* esrc, const int* edst,
                          int N, int E, int NE, hipStream_t stream) {
  const int tiles = (N + 15) / 16;
  gat_gemm128_f32_wmma<<<(tiles + 7) / 8, 256, 0, stream>>>(xin, W, h, N);
  gat_attn_prep<<<(N * 4 + 255) / 256, 256, 0, stream>>>(h, att_s, att_d,
                                                         as, ad, mm, sb, N);
  gat_edge_logit_max<<<(NE + 255) / 256, 256, 0, stream>>>(esrc, edst, as, ad,
                                                           elog, mm, E, NE);
  gat_edge_exp_sum<<<(NE + 255) / 256, 256, 0, stream>>>(edst, elog, mm, sb,
                                                         E, NE);
  hipMemsetAsync(outb, 0, (size_t)N * 128 * sizeof(float), stream);
  gat_edge_aggregate<<<(NE + 7) / 8, 256, 0, stream>>>(esrc, edst, elog, sb,
                                                       h, outb, E, NE);
}

extern "C" void kernel_launch(void* const* d_in, const int* in_sizes, int n_in,
                              void* d_out, int out_size, void* d_ws, size_t ws_size,
                              hipStream_t stream) {
  const float* x      = (const float*)d_in[0];
  const int*   eidx   = (const int*)d_in[1];
  // d_in[2] edge_weights: unused by reference (GATConv without edge_dim)
  const float* W1     = (const float*)d_in[3];
  const float* att_s1 = (const float*)d_in[4];
  const float* att_d1 = (const float*)d_in[5];
  const float* b1     = (const float*)d_in[6];
  const float* W2     = (const float*)d_in[7];
  const float* att_s2 = (const float*)d_in[8];
  const float* att_d2 = (const float*)d_in[9];
  const float* b2     = (const float*)d_in[10];
  float* out = (float*)d_out;

  const int N  = in_sizes[0] / 128;
  const int E  = in_sizes[1] / 2;
  const int NE = E + N;
  const int* esrc = eidx;          // edge_index[0]
  const int* edst = eidx + E;      // edge_index[1]

  // Workspace carve-up (all 256B aligned). Total ~136MB -> L2 resident.
  char* ws = (char*)d_ws;
  size_t off = 0;
  auto alloc = [&](size_t bytes) -> float* {
    float* p = (float*)(ws + off);
    off += (bytes + 255) & ~(size_t)255;
    return p;
  };
  float* hbuf = alloc((size_t)N * 128 * sizeof(float));  // h1, then h2
  float* obuf = alloc((size_t)N * 128 * sizeof(float));  // out1/h2in, then out2
  float* as   = alloc((size_t)N * 4 * sizeof(float));
  float* ad   = alloc((size_t)N * 4 * sizeof(float));
  float* mm   = alloc((size_t)N * 4 * sizeof(float));
  float* sb   = alloc((size_t)N * 4 * sizeof(float));
  float* elog = alloc((size_t)NE * 4 * sizeof(float));
  (void)ws_size; (void)n_in; (void)out_size;

  // Layer 1 (concat heads -> [N,128]), then elu(out + b1)
  gat_run_layer(x, W1, att_s1, att_d1, hbuf, obuf, as, ad, mm, sb, elog,
                esrc, edst, N, E, NE, stream);
  gat_elu_bias<<<((N * 128) + 255) / 256, 256, 0, stream>>>(obuf, b1, N * 128);

  // Layer 2: GEMM reads obuf -> hbuf; aggregation re-uses obuf as out2
  gat_run_layer(obuf, W2, att_s2, att_d2, hbuf, obuf, as, ad, mm, sb, elog,
                esrc, edst, N, E, NE, stream);

  // Mean over heads + b2 -> [N,32]
  gat_final_mean<<<((N * 32) + 255) / 256, 256, 0, stream>>>(obuf, b2, out, N);
}